// DAGNNLayer_38019050505085
// MI455X (gfx1250) — compile-verified
//
#include <hip/hip_runtime.h>

#define DIM 64

typedef float v2f __attribute__((ext_vector_type(2)));
typedef float v8f __attribute__((ext_vector_type(8)));

__global__ void zero_kernel(float* __restrict__ p, long n) {
    long i = (long)blockIdx.x * blockDim.x + threadIdx.x;
    if (i < n) p[i] = 0.0f;
}

__global__ void degree_kernel(const int* __restrict__ dst, float* __restrict__ deg, int E) {
    int e = blockIdx.x * blockDim.x + threadIdx.x;
    if (e < E) atomicAdd(&deg[dst[e]], 1.0f);
}

__global__ void norm_kernel(const float* __restrict__ deg, float* __restrict__ nrm, int N) {
    int i = blockIdx.x * blockDim.x + threadIdx.x;
    if (i < N) nrm[i] = rsqrtf(fmaxf(deg[i], 1.0f));
}

// Edge-parallel SpMM scatter: one wave32 per edge, 2 floats per lane.
__global__ __launch_bounds__(256)
void scatter_kernel(const float* __restrict__ g, const int* __restrict__ src,
                    const int* __restrict__ dst, float* __restrict__ acc, int E) {
    int lane = threadIdx.x & 31;
    int wib  = threadIdx.x >> 5;
    long e = (long)blockIdx.x * 8 + wib;
    if (e >= E) return;
    int s = src[e];
    int d = dst[e];
    float2 v = *(const float2*)(g + (long)s * DIM + lane * 2);
    float* ap = acc + (long)d * DIM + lane * 2;
    atomicAdd(ap, v.x);
    atomicAdd(ap + 1, v.y);
}

// Per-hop finish: h = raw * (pre_scale? norm : 1); dot = h . s  (via WMMA f32 16x16x4);
// out (+)= sigmoid(dot) * h; g_out = h * norm; optionally zero raw (the scatter accumulator).
// 256 threads = 8 waves; each wave owns a 16-node x 64-dim tile in LDS.
__global__ __launch_bounds__(256)
void finish_kernel(float* __restrict__ raw, const float* __restrict__ nrm,
                   const float* __restrict__ s_vec, float* __restrict__ g_out,
                   float* __restrict__ out, int N,
                   int pre_scale, int zero_src, int init_out) {
    __shared__ float tile[8 * 16 * DIM];   // 32 KB
    __shared__ float svals[DIM];
    __shared__ float sdot[8 * 16];

    const int lane = threadIdx.x & 31;
    const int w    = threadIdx.x >> 5;
    const int node0 = blockIdx.x * 128 + w * 16;
    float* t = &tile[w * 16 * DIM];

    if (threadIdx.x < DIM) svals[threadIdx.x] = s_vec[threadIdx.x];

    // Load 16x64 tile (coalesced float4), fuse pre-scale, emit g, zero source.
    #pragma unroll
    for (int i = 0; i < 8; ++i) {
        int f4 = i * 32 + lane;        // float4 index within tile [0,256)
        int f  = f4 * 4;               // flat float index [0,1024)
        int nl = f >> 6;               // local node 0..15
        int node = node0 + nl;
        int nodec = node < N ? node : N - 1;
        float4 r = *(const float4*)(raw + (long)nodec * DIM + (f & 63));
        float nm = nrm[nodec];
        float pre = pre_scale ? nm : 1.0f;
        float4 h = make_float4(r.x * pre, r.y * pre, r.z * pre, r.w * pre);
        ((float4*)t)[f4] = h;
        if (node < N) {
            float4 g4 = make_float4(h.x * nm, h.y * nm, h.z * nm, h.w * nm);
            *(float4*)(g_out + (long)node * DIM + (f & 63)) = g4;
            if (zero_src)
                *(float4*)(raw + (long)node * DIM + (f & 63)) = make_float4(0.f, 0.f, 0.f, 0.f);
        }
    }
    __syncthreads();

    // dot[m] = H_tile(16x64) . s  via 16 chained V_WMMA_F32_16X16X4_F32.
    // A 16x4 f32 layout: lanes 0-15 hold M=lane with {K0,K1} in 2 VGPRs; lanes 16-31 hold {K2,K3}.
    // B 4x16: column-uniform (= s chunk), filled with the mirrored K mapping -> permutation-safe dot.
    v8f c = {0.f, 0.f, 0.f, 0.f, 0.f, 0.f, 0.f, 0.f};
    const int row  = lane & 15;
    const int kofs = (lane < 16) ? 0 : 2;
    #pragma unroll
    for (int kc = 0; kc < 16; ++kc) {
        v2f a, b;
        a.x = t[row * DIM + kc * 4 + kofs];
        a.y = t[row * DIM + kc * 4 + kofs + 1];
        b.x = svals[kc * 4 + kofs];
        b.y = svals[kc * 4 + kofs + 1];
        c = __builtin_amdgcn_wmma_f32_16x16x4_f32(false, a, false, b, (short)0, c,
                                                  false, false);
    }

    // D layout: VGPR v, lanes 0-15 -> M=v ; lanes 16-31 -> M=v+8 (all columns equal here).
    if (lane == 0 || lane == 16) {
        int mbase = (lane == 0) ? 0 : 8;
        #pragma unroll
        for (int i = 0; i < 8; ++i) {
            float dsum = c[i];
            sdot[w * 16 + mbase + i] = 1.0f / (1.0f + __expf(-dsum));
        }
    }
    __syncthreads();

    // out[node] (+)= sigmoid(dot)*h ; lane handles dims {2*lane, 2*lane+1}.
    #pragma unroll 4
    for (int m = 0; m < 16; ++m) {
        int node = node0 + m;
        if (node >= N) break;
        float wgt = sdot[w * 16 + m];
        float2 hv = *(float2*)&t[m * DIM + lane * 2];
        float* op = out + (long)node * DIM + lane * 2;
        float2 o;
        if (init_out) {
            o.x = wgt * hv.x;
            o.y = wgt * hv.y;
        } else {
            float2 prev = *(float2*)op;
            o.x = prev.x + wgt * hv.x;
            o.y = prev.y + wgt * hv.y;
        }
        *(float2*)op = o;
    }
}

extern "C" void kernel_launch(void* const* d_in, const int* in_sizes, int n_in,
                              void* d_out, int out_size, void* d_ws, size_t ws_size,
                              hipStream_t stream) {
    const float* features = (const float*)d_in[0];
    const float* s_vec    = (const float*)d_in[1];
    const int*   src      = (const int*)d_in[2];
    const int*   dst      = (const int*)d_in[3];
    const int N = in_sizes[0] / DIM;
    const int E = in_sizes[2];
    const int K = 10;  // reference constant (python scalar k)

    // Carve workspace: deg[N], norm[N], g[N*64], acc[N*64]  (~52 MB)
    char* ws = (char*)d_ws;
    size_t off = 0;
    auto carve = [&](size_t bytes) {
        void* p = ws + off;
        off = (off + bytes + 255) & ~(size_t)255;
        return p;
    };
    float* deg = (float*)carve((size_t)N * sizeof(float));
    float* nrm = (float*)carve((size_t)N * sizeof(float));
    float* g   = (float*)carve((size_t)N * DIM * sizeof(float));
    float* acc = (float*)carve((size_t)N * DIM * sizeof(float));

    long nd = (long)N * DIM;
    zero_kernel<<<(unsigned)((N + 255) / 256), 256, 0, stream>>>(deg, N);
    zero_kernel<<<(unsigned)((nd + 255) / 256), 256, 0, stream>>>(acc, nd);
    degree_kernel<<<(E + 255) / 256, 256, 0, stream>>>(dst, deg, E);
    norm_kernel<<<(N + 255) / 256, 256, 0, stream>>>(deg, nrm, N);

    const int fblocks = (N + 127) / 128;
    // k=0 term: h0 = features (no pre-scale, don't touch input), g = features*norm, out = S0*h0
    finish_kernel<<<fblocks, 256, 0, stream>>>((float*)features, nrm, s_vec, g,
                                               (float*)d_out, N,
                                               /*pre_scale=*/0, /*zero_src=*/0, /*init_out=*/1);
    for (int hop = 0; hop < K; ++hop) {
        scatter_kernel<<<(E + 7) / 8, 256, 0, stream>>>(g, src, dst, acc, E);
        // h_k = acc*norm ; out += S_k*h_k ; g = h_k*norm ; acc reset to 0 for next hop
        finish_kernel<<<fblocks, 256, 0, stream>>>(acc, nrm, s_vec, g,
                                                   (float*)d_out, N,
                                                   /*pre_scale=*/1, /*zero_src=*/1, /*init_out=*/0);
    }
}